// Attention_10874857193481
// MI455X (gfx1250) — compile-verified
//
#include <hip/hip_runtime.h>

// ---- problem constants (match reference) ----
#define B_    16
#define H_    32
#define KVH_  8
#define G_    4
#define DH_   128
#define BS_   256     // paged block size
#define BPS_  16      // blocks per sequence
#define P_    8       // sequence partitions (flash-decoding)
#define PART_ 512     // keys per partition (P_*PART_ = 4096)
#define SCALE_ 0.08838834764831845f

typedef __attribute__((ext_vector_type(2))) float v2f;
typedef __attribute__((ext_vector_type(8))) float v8f;

__device__ __forceinline__ v8f wmma_f32(v2f a, v2f b, v8f c) {
  // D(16x16) = A(16x4) * B(4x16) + C, full fp32
  return __builtin_amdgcn_wmma_f32_16x16x4_f32(
      /*neg_a=*/false, a, /*neg_b=*/false, b,
      /*c_mod=*/(short)0, c, /*reuse_a=*/false, /*reuse_b=*/false);
}

// One wave (32 threads) per (b, kvh, partition).
// Layout facts used (CDNA5 ISA 7.12.2, 32-bit operands, wave32):
//   A 16x4 : lane L -> M = L%16 ; VGPR v -> K = (L/16)*2 + v
//   B 4x16 : lane L -> N = L%16 ; VGPR v -> K = (L/16)*2 + v
//   C/D    : lane L -> N = L%16 ; VGPR v -> M = v + 8*(L/16)
__global__ __launch_bounds__(32)
void Attention_decode_partial(const float* __restrict__ q,
                              const float* __restrict__ knew,
                              const float* __restrict__ vnew,
                              const float* __restrict__ kcache,
                              const float* __restrict__ vcache,
                              const int*   __restrict__ block_tables,
                              const int*   __restrict__ ctx_lens,
                              float* __restrict__ ws_acc,  // [B,KVH,P_,G_,DH_]
                              float* __restrict__ ws_m,    // [B,KVH,P_,G_]
                              float* __restrict__ ws_l)    // [B,KVH,P_,G_]
{
  const int lane = threadIdx.x;
  const int lo   = lane & 15;   // M/N index within tile
  const int hi   = lane >> 4;   // which half of the K-dim split

  const int part = blockIdx.x % P_;
  const int kvh  = (blockIdx.x / P_) % KVH_;
  const int b    = blockIdx.x / (P_ * KVH_);

  const int ctx    = ctx_lens[b];
  const int jstart = part * PART_;
  const int jend   = (jstart + PART_ < ctx) ? (jstart + PART_) : ctx;
  const int slot   = (b * KVH_ + kvh) * P_ + part;

  if (jstart >= jend) {
    // empty partition: neutral partials
    if (hi == 0 && lo < G_) { ws_m[slot * G_ + lo] = -1e30f; ws_l[slot * G_ + lo] = 0.0f; }
    for (int i = lane; i < G_ * DH_; i += 32) ws_acc[(size_t)slot * G_ * DH_ + i] = 0.0f;
    return;
  }

  // ---- Q fragments (B operand for score GEMM), pre-scaled; heads>=G are zero
  v2f qf[32];
  {
    const float* qb = q + ((size_t)b * H_ + (size_t)kvh * G_ + (lo < G_ ? lo : 0)) * DH_;
    for (int c = 0; c < 32; ++c) {
      const int d0 = c * 4 + hi * 2;
      v2f t;
      if (lo < G_) { t.x = qb[d0] * SCALE_; t.y = qb[d0 + 1] * SCALE_; }
      else         { t.x = 0.0f;            t.y = 0.0f; }
      qf[c] = t;
    }
  }

  const size_t kvstride = (size_t)KVH_ * DH_;           // floats per cache row
  const float* kq = knew + ((size_t)b * KVH_ + kvh) * DH_;
  const float* vq = vnew + ((size_t)b * KVH_ + kvh) * DH_;
  const int    newj = ctx - 1;                          // position of the appended token

  float m_run = -1e30f, l_run = 0.0f;
  v8f acc[8];
  for (int n = 0; n < 8; ++n) acc[n] = (v8f)0.0f;       // acc[n][v]: dim n*16 + v + 8*hi, head lo

  const int t0 = jstart >> 4;
  const int t1 = (jend + 15) >> 4;

  for (int t = t0; t < t1; ++t) {
    const int    blk     = block_tables[b * BPS_ + (t >> 4)];
    const size_t rowbase = (size_t)blk * BS_ + (size_t)((t & 15) << 4);

    // ---- scores: D(key x head) = K_tile(16x4 chunks) x Q^T ----
    const int   jk     = (t << 4) + lo;                 // key handled by this lane (A rows)
    const bool  patchk = (jk == newj);
    const float* kp    = kcache + (rowbase + lo) * kvstride + (size_t)kvh * DH_;
    v8f cs = (v8f)0.0f;
    for (int c = 0; c < 32; ++c) {
      const int d0 = c * 4 + hi * 2;
      v2f a;
      a.x = kp[d0]; a.y = kp[d0 + 1];
      if (patchk) { a.x = kq[d0]; a.y = kq[d0 + 1]; }   // new-token K
      cs = wmma_f32(a, qf[c], cs);
    }

    // ---- online softmax update (head = lo on this lane; keys v + 8*hi) ----
    float s[8];
    const int kb = (t << 4) + 8 * hi;
    for (int v = 0; v < 8; ++v) { const int j = kb + v; s[v] = (j < ctx) ? cs[v] : -1e30f; }

    float tmax = s[0];
    for (int v = 1; v < 8; ++v) tmax = fmaxf(tmax, s[v]);
    tmax = fmaxf(tmax, __shfl_xor(tmax, 16, 32));
    const float m_new = fmaxf(m_run, tmax);
    const float alpha = __expf(m_run - m_new);

    float p[8], psum = 0.0f;
    for (int v = 0; v < 8; ++v) { p[v] = __expf(s[v] - m_new); psum += p[v]; }
    psum += __shfl_xor(psum, 16, 32);
    l_run = l_run * alpha + psum;
    m_run = m_new;
    for (int n = 0; n < 8; ++n) acc[n] *= alpha;

    // ---- rearrange P into B-operand fragments (needs cross-half exchange) ----
    float sh[8];
    for (int v = 0; v < 8; ++v) sh[v] = __shfl_xor(p[v], 16, 32);
    v2f bf[4];
    if (hi) {
      bf[0].x = sh[2]; bf[0].y = sh[3];
      bf[1].x = sh[6]; bf[1].y = sh[7];
      bf[2].x = p[2];  bf[2].y = p[3];
      bf[3].x = p[6];  bf[3].y = p[7];
    } else {
      bf[0].x = p[0];  bf[0].y = p[1];
      bf[1].x = p[4];  bf[1].y = p[5];
      bf[2].x = sh[0]; bf[2].y = sh[1];
      bf[3].x = sh[4]; bf[3].y = sh[5];
    }

    // ---- output (transposed): D(dim x head) += V^T(16x4) x P(4x16) ----
    for (int n = 0; n < 8; ++n) {
      const int dim = n * 16 + lo;                      // A rows = dims
      for (int c2 = 0; c2 < 4; ++c2) {
        const int    kk = c2 * 4 + hi * 2;              // key chunk for this lane
        const int    j0 = (t << 4) + kk;
        const float* vp = vcache + (rowbase + kk) * kvstride + (size_t)kvh * DH_ + dim;
        v2f a;
        a.x = vp[0];
        a.y = vp[kvstride];
        if (j0     == newj) a.x = vq[dim];              // new-token V
        if (j0 + 1 == newj) a.y = vq[dim];
        acc[n] = wmma_f32(a, bf[c2], acc[n]);
      }
    }
  }

  // ---- write partials: acc[n][v] -> head lo, dim n*16 + v + 8*hi ----
  if (lo < G_) {
    const size_t abase = (size_t)slot * G_ * DH_ + (size_t)lo * DH_;
    for (int n = 0; n < 8; ++n)
      for (int v = 0; v < 8; ++v)
        ws_acc[abase + n * 16 + v + 8 * hi] = acc[n][v];
    if (hi == 0) { ws_m[slot * G_ + lo] = m_run; ws_l[slot * G_ + lo] = l_run; }
  }
}

// Merge the P_ partition partials: one block per (b, kvh), 128 threads = dims.
__global__ __launch_bounds__(128)
void Attention_decode_reduce(const float* __restrict__ ws_acc,
                             const float* __restrict__ ws_m,
                             const float* __restrict__ ws_l,
                             float* __restrict__ out)
{
  const int bk  = blockIdx.x;           // b*KVH_ + kvh
  const int b   = bk / KVH_;
  const int kvh = bk % KVH_;
  const int d   = threadIdx.x;          // 0..127

  for (int g = 0; g < G_; ++g) {
    float mv[P_], lv[P_];
    float M = -1e30f;
    for (int p = 0; p < P_; ++p) {
      mv[p] = ws_m[(bk * P_ + p) * G_ + g];
      lv[p] = ws_l[(bk * P_ + p) * G_ + g];
      M = fmaxf(M, mv[p]);
    }
    float L = 0.0f, o = 0.0f;
    for (int p = 0; p < P_; ++p) {
      const float w = __expf(mv[p] - M);
      L += lv[p] * w;
      o += ws_acc[((size_t)(bk * P_ + p) * G_ + g) * DH_ + d] * w;
    }
    out[((size_t)b * H_ + (size_t)kvh * G_ + g) * DH_ + d] = o / L;
  }
}

extern "C" void kernel_launch(void* const* d_in, const int* in_sizes, int n_in,
                              void* d_out, int out_size, void* d_ws, size_t ws_size,
                              hipStream_t stream) {
  const float* q        = (const float*)d_in[0];  // [16,32,128]
  const float* knew     = (const float*)d_in[1];  // [16,8,128]
  const float* vnew     = (const float*)d_in[2];  // [16,8,128]
  const float* kcache   = (const float*)d_in[3];  // [256,256,8,128]
  const float* vcache   = (const float*)d_in[4];  // [256,256,8,128]
  // d_in[5] = slot_mapping (unused: patching j==ctx-1 is equivalent, caches stay immutable)
  const int*   btables  = (const int*)d_in[6];    // [16,16]
  const int*   ctxlens  = (const int*)d_in[7];    // [16]
  float*       out      = (float*)d_out;          // [16,32,128]

  // workspace carve-up (~2.03 MB)
  const size_t ACC_ELEMS = (size_t)B_ * KVH_ * P_ * G_ * DH_;  // 524288 floats
  const size_t ML_ELEMS  = (size_t)B_ * KVH_ * P_ * G_;        // 4096 floats
  float* ws_acc = (float*)d_ws;
  float* ws_m   = ws_acc + ACC_ELEMS;
  float* ws_l   = ws_m + ML_ELEMS;
  (void)in_sizes; (void)n_in; (void)out_size; (void)ws_size;

  dim3 grid1(B_ * KVH_ * P_);   // 1024 single-wave workgroups
  dim3 blk1(32);
  Attention_decode_partial<<<grid1, blk1, 0, stream>>>(
      q, knew, vnew, kcache, vcache, btables, ctxlens, ws_acc, ws_m, ws_l);

  dim3 grid2(B_ * KVH_);        // 128 workgroups
  dim3 blk2(128);
  Attention_decode_reduce<<<grid2, blk2, 0, stream>>>(ws_acc, ws_m, ws_l, out);
}